// PhaseLinear_46145128628239
// MI455X (gfx1250) — compile-verified
//
#include <hip/hip_runtime.h>
#include <hip/hip_bf16.h>

// ---------------------------------------------------------------------------
// PhaseLinear on MI455X (gfx1250):
//   out[b] = sum_p ceff[b,p] * (W_p @ x[b] + bias_p)
// Formulated as 4 concurrent f32 WMMA GEMM accumulations (16x16x4 f32 tiles)
// with the per-row spline-coefficient combine folded into the epilogue.
// ---------------------------------------------------------------------------

#define B_TOTAL 2048
#define IN_DIM  256
#define OUT_DIM 256

typedef __attribute__((ext_vector_type(2))) float v2f;
typedef __attribute__((ext_vector_type(8))) float v8f;

// ---------------------------------------------------------------------------
// Kernel 1: per-batch spline coefficients.
//   bucket = #(bounds[i] < phase) for bounds = {0, pi/2, pi, 3pi/2}  (phase < 1.5pi)
//   t      = (phase - bucket*pi/2) / (pi/2)
//   coef   = [t^3, t^2, t, 1] @ basis
//   ceff[(k + bucket + 3) & 3] = coef[k]   (CP_IDX is a rotation)
// ---------------------------------------------------------------------------
__global__ void phase_coef_kernel(const float* __restrict__ phase,
                                  const float* __restrict__ basis,
                                  float* __restrict__ ceff, int n) {
    int b = blockIdx.x * blockDim.x + threadIdx.x;
    if (b >= n) return;
    const float HPI = 1.57079632679489662f;   // pi/2
    float ph = phase[b];
    int idx = (ph > 0.0f) + (ph > HPI) + (ph > 2.0f * HPI);
    int bucket = idx > 3 ? 3 : idx;
    float t  = (ph - (float)bucket * HPI) / HPI;
    float t2 = t * t;
    float t3 = t2 * t;
    float coef[4];
#pragma unroll
    for (int j = 0; j < 4; ++j)
        coef[j] = t3 * basis[0 * 4 + j] + t2 * basis[1 * 4 + j]
                +  t * basis[2 * 4 + j] +      basis[3 * 4 + j];
    float ce[4];
#pragma unroll
    for (int k = 0; k < 4; ++k)
        ce[(k + bucket + 3) & 3] = coef[k];
    float4 o;
    o.x = ce[0]; o.y = ce[1]; o.z = ce[2]; o.w = ce[3];
    ((float4*)ceff)[b] = o;
}

// ---------------------------------------------------------------------------
// Kernel 2: GEMM + combine.
//   Block = 256 threads = 8 waves. Block owns one 16-row M tile (staged in LDS).
//   Wave w owns N tile [nbase, nbase+16). Four accumulator fragments (one per
//   control point) advance together over K in steps of 4 via
//   v_wmma_f32_16x16x4_f32, giving 4 independent XDL dependency chains.
//
//   f32 A 16x4 layout (ISA 7.12.2): lane L holds row M=L&15; VGPR0/1 hold
//   K = (L>=16 ? 2 : 0) + {0,1}.  B 4x16 mirrors it: VGPR0/1 lanes 0-15 hold
//   K=0/1 for N=0..15, lanes 16-31 hold K=2/3.
//   C/D: VGPR r, lanes 0-15 -> M=r; lanes 16-31 -> M=r+8; N = lane&15.
// ---------------------------------------------------------------------------
__global__ __launch_bounds__(256)
void phase_linear_gemm_kernel(const float* __restrict__ x,
                              const float* __restrict__ weights,
                              const float* __restrict__ biases,
                              const float* __restrict__ ceff,
                              float* __restrict__ out) {
    __shared__ float lds_x[16 * IN_DIM];   // 16 KB: X tile, shared by 8 waves
    __shared__ float lds_c[16 * 4];        // ceff for the 16 rows of this tile

    const int tid   = threadIdx.x;
    const int lane  = tid & 31;
    const int wave  = tid >> 5;
    const int mbase = blockIdx.x * 16;
    const int nbase = blockIdx.y * 128 + wave * 16;

    // Stage X tile (coalesced float4 copies) and ceff rows into LDS.
    {
        const float4* src = (const float4*)(x + (size_t)mbase * IN_DIM);
        float4* dst = (float4*)lds_x;
#pragma unroll
        for (int i = tid; i < 16 * IN_DIM / 4; i += 256) dst[i] = src[i];
    }
    if (tid < 64) lds_c[tid] = ceff[mbase * 4 + tid];
    __syncthreads();

    const int row  = lane & 15;           // M row (A) / N column (B,C,D)
    const int kb   = (lane >> 4) << 1;    // K sub-offset: 0 for lanes 0-15, 2 else
    const int ocol = nbase + row;         // this lane's output channel

    // Per-lane bias for each control point (column is fixed for the wave).
    const float bia0 = biases[0 * OUT_DIM + ocol];
    const float bia1 = biases[1 * OUT_DIM + ocol];
    const float bia2 = biases[2 * OUT_DIM + ocol];
    const float bia3 = biases[3 * OUT_DIM + ocol];

    // B operand base pointers: weights[p][ocol][*]
    const float* wp0 = weights + ((size_t)(0 * OUT_DIM + ocol)) * IN_DIM + kb;
    const float* wp1 = weights + ((size_t)(1 * OUT_DIM + ocol)) * IN_DIM + kb;
    const float* wp2 = weights + ((size_t)(2 * OUT_DIM + ocol)) * IN_DIM + kb;
    const float* wp3 = weights + ((size_t)(3 * OUT_DIM + ocol)) * IN_DIM + kb;

    const float* xrow = lds_x + row * IN_DIM + kb;

    v8f acc0 = {}, acc1 = {}, acc2 = {}, acc3 = {};

#pragma unroll 4
    for (int k = 0; k < IN_DIM; k += 4) {
        v2f a  = *(const v2f*)(xrow + k);       // 8B-aligned LDS read
        v2f b0 = *(const v2f*)(wp0 + k);        // 8B-aligned global reads (L2-hot)
        v2f b1 = *(const v2f*)(wp1 + k);
        v2f b2 = *(const v2f*)(wp2 + k);
        v2f b3 = *(const v2f*)(wp3 + k);
        acc0 = __builtin_amdgcn_wmma_f32_16x16x4_f32(false, a, false, b0,
                                                     (short)0, acc0, false, false);
        acc1 = __builtin_amdgcn_wmma_f32_16x16x4_f32(false, a, false, b1,
                                                     (short)0, acc1, false, false);
        acc2 = __builtin_amdgcn_wmma_f32_16x16x4_f32(false, a, false, b2,
                                                     (short)0, acc2, false, false);
        acc3 = __builtin_amdgcn_wmma_f32_16x16x4_f32(false, a, false, b3,
                                                     (short)0, acc3, false, false);
    }

    // Epilogue: out[b, ocol] = sum_p ceff[b][p] * (acc_p + bias_p)
    const int mrow_off = (lane >> 4) * 8;
#pragma unroll
    for (int r = 0; r < 8; ++r) {
        const int ml = r + mrow_off;          // local M row of this fragment elem
        const float c0 = lds_c[ml * 4 + 0];
        const float c1 = lds_c[ml * 4 + 1];
        const float c2 = lds_c[ml * 4 + 2];
        const float c3 = lds_c[ml * 4 + 3];
        const float v = c0 * (acc0[r] + bia0) + c1 * (acc1[r] + bia1)
                      + c2 * (acc2[r] + bia2) + c3 * (acc3[r] + bia3);
        out[(size_t)(mbase + ml) * OUT_DIM + ocol] = v;
    }
}

// ---------------------------------------------------------------------------
// Launch wrapper.
// Inputs (setup_inputs order): input[2048,256] f32, phase[2048] f32,
// weights[4,256,256] f32, biases[4,256] f32, basis[4,4] f32.
// d_ws: 2048*4 floats (32 KB) for ceff.
// ---------------------------------------------------------------------------
extern "C" void kernel_launch(void* const* d_in, const int* in_sizes, int n_in,
                              void* d_out, int out_size, void* d_ws, size_t ws_size,
                              hipStream_t stream) {
    const float* x       = (const float*)d_in[0];
    const float* phase   = (const float*)d_in[1];
    const float* weights = (const float*)d_in[2];
    const float* biases  = (const float*)d_in[3];
    const float* basis   = (const float*)d_in[4];
    float* out  = (float*)d_out;
    float* ceff = (float*)d_ws;

    const int nB = in_sizes[1];   // 2048

    phase_coef_kernel<<<(nB + 255) / 256, 256, 0, stream>>>(phase, basis, ceff, nB);

    dim3 grid(nB / 16, OUT_DIM / 128);   // 128 M-tiles x 2 N-halves
    phase_linear_gemm_kernel<<<grid, 256, 0, stream>>>(x, weights, biases, ceff, out);
}